// AugmentModel_24979529794182
// MI455X (gfx1250) — compile-verified
//
#include <hip/hip_runtime.h>
#include <hip/hip_bf16.h>
#include <math.h>

// Problem constants (match reference)
#define KK 7
#define BB 16
#define CC 3
#define HH 256
#define WW 256
#define NN 50000
#define UPH 1024
#define UPW 1024
#define TILES 8                      // output row-bands per (b,c) plane
#define ROWS_PER_TILE (HH / TILES)   // 32
#define PLANE_ELEMS (HH * WW)        // 65536 floats = 256 KB (fits 320 KB LDS)
#define PLANE_BYTES (PLANE_ELEMS * 4)
#define TDM_ROWS 128                 // rows 0..127 staged by the Tensor Data Mover
#define ASYNC_BYTE0 (TDM_ROWS * WW * 4)  // rows 128..255 staged by per-lane async DMA

typedef unsigned int u32x4 __attribute__((ext_vector_type(4)));
typedef int i32x4 __attribute__((ext_vector_type(4)));
typedef int i32x8 __attribute__((ext_vector_type(8)));

__device__ __forceinline__ float sigmoidf_(float v) { return 1.0f / (1.0f + expf(-v)); }

// ---------------------------------------------------------------------------
// Kernel 1: build per-batch inverse affine matrices (16 tiny 3x3 chains).
// ---------------------------------------------------------------------------
__global__ void make_mats_kernel(const float* __restrict__ paramP,
                                 const float* __restrict__ paramM,
                                 const float* __restrict__ gumbel_u,
                                 const float* __restrict__ eps,
                                 const int* __restrict__ idx,
                                 float* __restrict__ mats) {
  int b = threadIdx.x;
  if (b >= BB) return;
  int j = idx[b];
  float sP[KK], sM[KK];
#pragma unroll
  for (int k = 0; k < KK; ++k) {
    float pP = sigmoidf_(paramP[k * NN + j]);
    int r = k * BB + b;
    // Gumbel-softmax hard sample: argmax over (log p + g0, log(1-p) + g1).
    float g0 = -logf(-logf(gumbel_u[2 * r + 0]));
    float g1 = -logf(-logf(gumbel_u[2 * r + 1]));
    float l0 = logf(pP) + g0;
    float l1 = logf(1.0f - pP) + g1;
    sP[k] = (l0 >= l1) ? 1.0f : 0.0f;   // jnp.argmax picks index 0 on tie
    sM[k] = sigmoidf_(paramM[k * NN + j]) * eps[r];
  }
  float ang = sP[0] * sM[0] * 30.0f;
  float rad = ang * (3.14159265358979323846f / 180.0f);
  float ca = cosf(rad), sa = sinf(rad);
  float sx = 1.0f + sP[5] * sM[5] * 0.5f;
  float sy = 1.0f + sP[6] * sM[6] * 0.5f;
  float sr00 = ca * sx, sr01 = sa * sy;
  float sr10 = -sa * sx, sr11 = ca * sy;
  const float cx = (float)(WW / 2), cy = (float)(HH / 2);  // 128,128 (orig H,W per ref)
  float tx = cx - (sr00 * cx + sr01 * cy);
  float ty = cy - (sr10 * cx + sr11 * cy);
  float Tx = (float)WW * sP[1] * sM[1] * 0.45f;
  float Ty = (float)HH * sP[2] * sM[2] * 0.45f;
  float s01 = sP[3] * sM[3] * 0.3f;
  float s10 = sP[4] * sM[4] * 0.3f;
  // TR = T @ R ; M = S @ TR
  float TR00 = sr00, TR01 = sr01, TR02 = tx + Tx;
  float TR10 = sr10, TR11 = sr11, TR12 = ty + Ty;
  float M00 = TR00 + s01 * TR10;
  float M01 = TR01 + s01 * TR11;
  float M02 = TR02 + s01 * TR12;
  float M10 = s10 * TR00 + TR10;
  float M11 = s10 * TR01 + TR11;
  float M12 = s10 * TR02 + TR12;
  // Affine inverse (last row 0,0,1 -> perspective divide is identity)
  float det = M00 * M11 - M01 * M10;
  float rdet = 1.0f / det;
  float i00 = M11 * rdet, i01 = -M01 * rdet;
  float i10 = -M10 * rdet, i11 = M00 * rdet;
  float i02 = -(i00 * M02 + i01 * M12);
  float i12 = -(i10 * M02 + i11 * M12);
  mats[b * 8 + 0] = i00;
  mats[b * 8 + 1] = i01;
  mats[b * 8 + 2] = i02;
  mats[b * 8 + 3] = i10;
  mats[b * 8 + 4] = i11;
  mats[b * 8 + 5] = i12;
  mats[b * 8 + 6] = 0.0f;
  mats[b * 8 + 7] = 0.0f;
}

// Bilinear tap of the 256x256 LDS-resident plane at the (virtual) 1024-res
// upsample grid position (Yu, Xu): src = coord*0.25 - 0.375, clipped.
__device__ __forceinline__ float up_sample(const float* plane, int Yu, int Xu) {
  float sy = fmaf((float)Yu, 0.25f, -0.375f);
  float sx = fmaf((float)Xu, 0.25f, -0.375f);
  sy = fminf(fmaxf(sy, 0.0f), (float)(HH - 1));
  sx = fminf(fmaxf(sx, 0.0f), (float)(WW - 1));
  float fy0 = floorf(sy), fx0 = floorf(sx);
  float wy = sy - fy0, wx = sx - fx0;
  int y0 = (int)fy0, x0 = (int)fx0;
  int y1 = min(y0 + 1, HH - 1), x1 = min(x0 + 1, WW - 1);
  const float* r0 = plane + y0 * WW;
  const float* r1 = plane + y1 * WW;
  float a = r0[x0], bq = r0[x1], cq = r1[x0], d = r1[x1];
  return a * (1.0f - wx) * (1.0f - wy) + bq * wx * (1.0f - wy) +
         cq * (1.0f - wx) * wy + d * wx * wy;
}

// ---------------------------------------------------------------------------
// Kernel 2: fully fused  resize(4x) -> warp(Minv) -> resize(1/4).
// Plane staged into LDS by TDM DMA (rows 0..127, one tensor_load_to_lds) +
// per-lane async global->LDS (rows 128..255), then all gathers are LDS-local.
// ---------------------------------------------------------------------------
__global__ __launch_bounds__(1024, 1)
void warp_fused_kernel(const float* __restrict__ x,
                       const float* __restrict__ mats,
                       float* __restrict__ out) {
  extern __shared__ float plane[];   // 65536 floats = 256 KB
  const int tile = blockIdx.x;       // 0..TILES-1 row band
  const int c = blockIdx.y;
  const int b = blockIdx.z;
  const int tid = threadIdx.x;

  const float* src = x + (size_t)(b * CC + c) * PLANE_ELEMS;
  unsigned lds_base = (unsigned)(size_t)(void*)plane;  // low 32b = LDS byte offset
  unsigned long long gbase = (unsigned long long)(const void*)src;

  // ---- (a) Tensor Data Mover: DMA rows 0..127 (256x128 f32 tile) to LDS. ----
  // Issued by wave 0 only (wave-uniform branch: TDM ignores EXEC, so it must
  // be branched over, not exec-masked). Tracked by TENSORcnt.
  if (tid < 32) {
    u32x4 g0;
    g0.x = 1u;                         // count=1 valid descriptor; not restore/gather
    g0.y = lds_base;                   // lds_addr [63:32]
    g0.z = (unsigned)gbase;            // global_addr [95:64]
    g0.w = (unsigned)((gbase >> 32) & 0x01FFFFFFull) | 0x80000000u;  // addr[56:32] | type=2
    i32x8 g1;
    g1[0] = (int)(2u << 16);           // workgroup_mask=0, data_size=2 (4 bytes)
    g1[1] = (int)((unsigned)WW << 16); // tensor_dim0 = 256 (low 16)
    g1[2] = (int)((unsigned)HH << 16); // tensor_dim0 hi=0 | tensor_dim1 = 256 (low 16)
    g1[3] = (int)((unsigned)WW << 16); // tensor_dim1 hi=0 | tile_dim0 = 256
    g1[4] = (int)(unsigned)TDM_ROWS;   // tile_dim1 = 128, tile_dim2 = 0 (2D tile)
    g1[5] = (int)(unsigned)WW;         // tensor_dim0_stride = 256 (low 32)
    g1[6] = 0;                         // stride0 hi | stride1 lo (unused, tile_dim2=0)
    g1[7] = 0;
    i32x4 gz = {0, 0, 0, 0};
#if defined(__clang_major__) && __clang_major__ >= 23
    i32x8 gz8 = {0, 0, 0, 0, 0, 0, 0, 0};
    __builtin_amdgcn_tensor_load_to_lds(g0, g1, gz, gz, gz8, 0);
#else
    __builtin_amdgcn_tensor_load_to_lds(g0, g1, gz, gz, 0);
#endif
  }

  // ---- (b) Per-lane async DMA: rows 128..255 (128 KB). ASYNCcnt-tracked. ----
  // 1024 threads x 8 ops x 16 B = 128 KB, overlapping with the TDM transfer.
#pragma unroll
  for (int i = 0; i < (PLANE_ELEMS / 2) / 4 / 1024; ++i) {
    unsigned byteoff = (unsigned)(ASYNC_BYTE0 + (i * 1024 + tid) * 16);
    unsigned ldsaddr = lds_base + byteoff;
    asm volatile("global_load_async_to_lds_b128 %0, %1, %2"
                 :
                 : "v"(ldsaddr), "v"(byteoff), "s"(gbase)
                 : "memory");
  }
  asm volatile("s_wait_asynccnt 0" ::: "memory");
  if (tid < 32) {
    __builtin_amdgcn_s_wait_tensorcnt(0);
  }
  __syncthreads();

  // Per-batch inverse affine (uniform across block -> scalar loads).
  const float i00 = mats[b * 8 + 0];
  const float i01 = mats[b * 8 + 1];
  const float i02 = mats[b * 8 + 2];
  const float i10 = mats[b * 8 + 3];
  const float i11 = mats[b * 8 + 4];
  const float i12 = mats[b * 8 + 5];

  const int pix_per_tile = ROWS_PER_TILE * WW;  // 8192
#pragma unroll 1
  for (int jj = 0; jj < pix_per_tile / 1024; ++jj) {
    int px = jj * 1024 + tid;
    int yl = px >> 8;           // /WW
    int xo = px & (WW - 1);
    int yo = tile * ROWS_PER_TILE + yl;

    // Down-resize source block in 1024-space: rows/cols {4y+1, 4y+2}, w=0.5
    // exactly -> plain 0.25 average of the 2x2 warped block.
    int Yb = 4 * yo + 1;
    int Xb = 4 * xo + 1;

    float acc = 0.0f;
#pragma unroll
    for (int dy = 0; dy < 2; ++dy) {
#pragma unroll
      for (int dx = 0; dx < 2; ++dx) {
        float X = (float)(Xb + dx);
        float Y = (float)(Yb + dy);
        // Inverse warp (affine; homogeneous w == 1).
        float fx = fmaf(i00, X, fmaf(i01, Y, i02));
        float fy = fmaf(i10, X, fmaf(i11, Y, i12));
        fx = fminf(fmaxf(fx, 0.0f), (float)(UPW - 1));
        fy = fminf(fmaxf(fy, 0.0f), (float)(UPH - 1));
        float fx0 = floorf(fx), fy0 = floorf(fy);
        float wx = fx - fx0, wy = fy - fy0;
        int ix0 = (int)fx0, iy0 = (int)fy0;
        int ix1 = min(ix0 + 1, UPW - 1);
        int iy1 = min(iy0 + 1, UPH - 1);
        float v00 = up_sample(plane, iy0, ix0);
        float v01 = up_sample(plane, iy0, ix1);
        float v10 = up_sample(plane, iy1, ix0);
        float v11 = up_sample(plane, iy1, ix1);
        acc += v00 * (1.0f - wx) * (1.0f - wy) + v01 * wx * (1.0f - wy) +
               v10 * (1.0f - wx) * wy + v11 * wx * wy;
      }
    }
    out[((size_t)(b * CC + c) * HH + yo) * WW + xo] = acc * 0.25f;
  }
}

extern "C" void kernel_launch(void* const* d_in, const int* in_sizes, int n_in,
                              void* d_out, int out_size, void* d_ws, size_t ws_size,
                              hipStream_t stream) {
  const float* x      = (const float*)d_in[0];
  const float* paramP = (const float*)d_in[1];
  const float* paramM = (const float*)d_in[2];
  const float* gumbel = (const float*)d_in[3];
  const float* eps    = (const float*)d_in[4];
  const int*   idx    = (const int*)d_in[5];
  float* out  = (float*)d_out;
  float* mats = (float*)d_ws;   // 16 * 8 floats

  make_mats_kernel<<<dim3(1), dim3(32), 0, stream>>>(paramP, paramM, gumbel,
                                                     eps, idx, mats);

  // 256 KB dynamic LDS (> default cap); attribute set is idempotent/deterministic.
  (void)hipFuncSetAttribute((const void*)warp_fused_kernel,
                            hipFuncAttributeMaxDynamicSharedMemorySize,
                            PLANE_BYTES);

  dim3 grid(TILES, CC, BB);     // 8 x 3 x 16 = 384 workgroups, 1 per WGP (LDS-bound)
  warp_fused_kernel<<<grid, dim3(1024), PLANE_BYTES, stream>>>(x, mats, out);
}